// NonMaxSupression_36180804501999
// MI455X (gfx1250) — compile-verified
//
#include <hip/hip_runtime.h>
#include <stdint.h>

#ifndef __has_builtin
#define __has_builtin(x) 0
#endif

// Problem geometry (fixed by the reference): [1,1,2048,4096] f32.
#define NMS_H 2048
#define NMS_W 4096

// Tile: 128 x 8 outputs per 256-thread block (8 wave32s). Each thread does 4
// consecutive pixels with 128-bit traffic on ori/out.
#define TILE_W 128
#define TILE_H 8
// LDS mag tile with halo, covering x-rel [-4, 131] so the [-1, 128] halo is
// inside and every 16B chunk is either fully in-image or fully out (W % 4 == 0).
#define LROW   136                 // floats per LDS row (34 float4 chunks)
#define LROWS  (TILE_H + 2)        // 10 rows: y-rel [-1, 8]
#define NCHUNK (LROWS * 34)        // 340 float4 async transfers per tile

typedef float v4f __attribute__((ext_vector_type(4)));
typedef int   v4i __attribute__((vector_size(16)));
typedef __attribute__((address_space(1))) v4i GV4;   // global int4*
typedef __attribute__((address_space(3))) v4i LV4;   // lds int4*

static __device__ __forceinline__ void async_tile_load_b128(const float* gp, float* lp) {
#if __has_builtin(__builtin_amdgcn_global_load_async_to_lds_b128)
    // (global src, lds dst, imm offset, imm cpol)
    __builtin_amdgcn_global_load_async_to_lds_b128((GV4*)gp, (LV4*)lp, 0, 0);
#else
    asm volatile("global_load_async_to_lds_b128 %0, %1, off"
                 :: "v"((uint32_t)(uintptr_t)lp), "v"((uint64_t)(uintptr_t)gp)
                 : "memory");
#endif
}

static __device__ __forceinline__ void wait_async0() {
#if __has_builtin(__builtin_amdgcn_s_wait_asynccnt)
    __builtin_amdgcn_s_wait_asynccnt(0);
#else
    asm volatile("s_wait_asynccnt 0" ::: "memory");
#endif
}

__global__ __launch_bounds__(256)
void nms_tile_kernel(const float* __restrict__ mag,
                     const float* __restrict__ ori,
                     const float* __restrict__ weight,  // [8][1][3][3] flat = 72
                     const float* __restrict__ bias,    // [8]
                     float* __restrict__ out)
{
    __shared__ __align__(16) float smag[LROWS * LROW]; // 10*136*4 = 5440 B
    __shared__ float sw[72];
    __shared__ float sb[8];

    const int tid = threadIdx.x;
    const int gx0 = blockIdx.x * TILE_W;
    const int gy0 = blockIdx.y * TILE_H;

    if (tid < 72) sw[tid] = weight[tid];
    if (tid < 8)  sb[tid] = bias[tid];

    // Stage mag tile (+halo) into LDS via async B128 transfers; zero-fill the
    // fully out-of-image chunks (SAME zero padding).
    for (int i = tid; i < NCHUNK; i += 256) {
        const int r  = i / 34;
        const int cc = i - r * 34;
        const int gy = gy0 + r - 1;
        const int gx = gx0 + cc * 4 - 4;
        float* lp = &smag[r * LROW + cc * 4];
        if ((unsigned)gy < (unsigned)NMS_H && (unsigned)gx < (unsigned)NMS_W) {
            async_tile_load_b128(&mag[gy * NMS_W + gx], lp);
        } else {
            v4f z = {0.f, 0.f, 0.f, 0.f};
            *(v4f*)lp = z;
        }
    }
    wait_async0();      // my wave's async LDS writes are done
    __syncthreads();    // everyone's LDS writes (ds zero-fills + asyncs) visible

    // Each thread: 4 consecutive pixels of one tile row.
    const int r   = tid >> 5;          // 0..7
    const int c4  = (tid & 31) << 2;   // 0,4,...,124
    const int gidx = (gy0 + r) * NMS_W + (gx0 + c4);

    const v4f o4 = __builtin_nontemporal_load((const v4f*)(ori + gidx));
    const float oa[4] = {o4.x, o4.y, o4.z, o4.w};

    // Center of pixel p is smag[(r+1)*LROW + c4 + 4 + p].
    const float* crow = &smag[(r + 1) * LROW + c4 + 4];

    float ra[4];
#pragma unroll
    for (int p = 0; p < 4; ++p) {
        const float o = oa[p];
        const int pos = ((int)(o / 45.0f)) & 7;   // o >= 0: trunc == floor
        const int neg = (pos + 4) & 7;            // == floor((o/45 + 4) mod 8)
        const float* wp = &sw[pos * 9];           // 8 variants -> distinct LDS banks
        const float* wn = &sw[neg * 9];
        float cp = sb[pos];
        float cn = sb[neg];
#pragma unroll
        for (int ky = 0; ky < 3; ++ky) {
#pragma unroll
            for (int kx = 0; kx < 3; ++kx) {
                const float d = crow[(ky - 1) * LROW + (p + kx - 1)];
                cp = fmaf(wp[ky * 3 + kx], d, cp);
                cn = fmaf(wn[ky * 3 + kx], d, cn);
            }
        }
        const float m = crow[p];
        ra[p] = (fminf(cp, cn) > 0.0f) ? m : 0.0f;
    }

    v4f res;
    res.x = ra[0]; res.y = ra[1]; res.z = ra[2]; res.w = ra[3];
    __builtin_nontemporal_store(res, (v4f*)(out + gidx));
}

extern "C" void kernel_launch(void* const* d_in, const int* in_sizes, int n_in,
                              void* d_out, int out_size, void* d_ws, size_t ws_size,
                              hipStream_t stream) {
    (void)in_sizes; (void)n_in; (void)out_size; (void)d_ws; (void)ws_size;
    const float* mag    = (const float*)d_in[0];
    const float* ori    = (const float*)d_in[1];
    const float* weight = (const float*)d_in[2];
    const float* bias   = (const float*)d_in[3];
    float* out = (float*)d_out;

    dim3 grid(NMS_W / TILE_W, NMS_H / TILE_H);   // 32 x 256 blocks
    nms_tile_kernel<<<grid, dim3(256), 0, stream>>>(mag, ori, weight, bias, out);
}